// GatingNetwork_49675591745735
// MI455X (gfx1250) — compile-verified
//
#include <hip/hip_runtime.h>
#include <math.h>

typedef __attribute__((ext_vector_type(2))) float v2f;
typedef __attribute__((ext_vector_type(8))) float v8f;

#define D_MODEL 2048
#define NEXP    64
#define TPW     16            // tokens per wave (WMMA M)
#define WPB     8             // waves per block
#define TPB     (TPW * WPB)   // 128 tokens per block
#define LPAD    68            // padded expert stride in LDS (bank-conflict-free)

__global__ __launch_bounds__(256) void gating_fused(
    const float* __restrict__ x,     // [n_tokens, D_MODEL]
    const float* __restrict__ gw,    // [NEXP, D_MODEL]
    const float* __restrict__ gb,    // [NEXP]
    float* __restrict__ out,         // topk_w [n,2] ++ topk_idx [n,2] ++ weights [n,64]
    int n_tokens)
{
    __shared__ float sl[WPB][TPW][LPAD];

    const int tid  = threadIdx.x;
    const int wave = tid >> 5;
    const int lane = tid & 31;
    const int half = lane >> 4;   // 0: K=k..k+1, 1: K=k+2..k+3 (fp32 WMMA frag layout)
    const int l16  = lane & 15;   // A: row M ; B: col N

    const int tok0 = blockIdx.x * TPB + wave * TPW;

    // A fragment source: x row (tok0 + l16), starting at K sub-offset 2*half
    const float* ax = x + (size_t)(tok0 + l16) * D_MODEL + 2 * half;
    // B fragment sources: 4 expert groups of 16, expert = g*16 + l16
    const float* bw0 = gw + (size_t)( 0 + l16) * D_MODEL + 2 * half;
    const float* bw1 = gw + (size_t)(16 + l16) * D_MODEL + 2 * half;
    const float* bw2 = gw + (size_t)(32 + l16) * D_MODEL + 2 * half;
    const float* bw3 = gw + (size_t)(48 + l16) * D_MODEL + 2 * half;

    v8f accA0 = {}, accA1 = {}, accA2 = {}, accA3 = {};
    v8f accB0 = {}, accB1 = {}, accB2 = {}, accB3 = {};

    // K loop: two independent WMMA chains per expert group (8 chains total)
    for (int k = 0; k < D_MODEL; k += 8) {
        v2f a0 = *(const v2f*)(ax + k);
        v2f a1 = *(const v2f*)(ax + k + 4);

        v2f b00 = *(const v2f*)(bw0 + k);
        v2f b01 = *(const v2f*)(bw0 + k + 4);
        v2f b10 = *(const v2f*)(bw1 + k);
        v2f b11 = *(const v2f*)(bw1 + k + 4);
        v2f b20 = *(const v2f*)(bw2 + k);
        v2f b21 = *(const v2f*)(bw2 + k + 4);
        v2f b30 = *(const v2f*)(bw3 + k);
        v2f b31 = *(const v2f*)(bw3 + k + 4);

        accA0 = __builtin_amdgcn_wmma_f32_16x16x4_f32(false, a0, false, b00, (short)0, accA0, false, false);
        accA1 = __builtin_amdgcn_wmma_f32_16x16x4_f32(false, a0, false, b10, (short)0, accA1, false, false);
        accA2 = __builtin_amdgcn_wmma_f32_16x16x4_f32(false, a0, false, b20, (short)0, accA2, false, false);
        accA3 = __builtin_amdgcn_wmma_f32_16x16x4_f32(false, a0, false, b30, (short)0, accA3, false, false);
        accB0 = __builtin_amdgcn_wmma_f32_16x16x4_f32(false, a1, false, b01, (short)0, accB0, false, false);
        accB1 = __builtin_amdgcn_wmma_f32_16x16x4_f32(false, a1, false, b11, (short)0, accB1, false, false);
        accB2 = __builtin_amdgcn_wmma_f32_16x16x4_f32(false, a1, false, b21, (short)0, accB2, false, false);
        accB3 = __builtin_amdgcn_wmma_f32_16x16x4_f32(false, a1, false, b31, (short)0, accB3, false, false);
    }

    // Spill logits to LDS. C/D layout: lane = N (expert%16), VGPR j = M (token row),
    // lanes 16..31 hold rows M=8..15.
    {
        v8f s0 = accA0 + accB0;
        v8f s1 = accA1 + accB1;
        v8f s2 = accA2 + accB2;
        v8f s3 = accA3 + accB3;
        #pragma unroll
        for (int j = 0; j < 8; ++j) {
            const int m = j + 8 * half;
            sl[wave][m][ 0 + l16] = s0[j];
            sl[wave][m][16 + l16] = s1[j];
            sl[wave][m][32 + l16] = s2[j];
            sl[wave][m][48 + l16] = s3[j];
        }
    }
    __syncthreads();

    // Softmax + top-2: one thread per token (threads 0..127)
    if (tid < TPB) {
        const int t = blockIdx.x * TPB + tid;
        if (t < n_tokens) {
            const float* lr = &sl[tid >> 4][tid & 15][0];

            float mx = -3.402823466e38f;
            #pragma unroll 8
            for (int e = 0; e < NEXP; ++e) {
                float v = lr[e] + gb[e];
                mx = fmaxf(mx, v);
            }

            float sum = 0.0f;
            #pragma unroll 8
            for (int e = 0; e < NEXP; ++e)
                sum += __expf(lr[e] + gb[e] - mx);
            const float inv = 1.0f / sum;

            float w1 = -1.0f, w2 = -1.0f;
            int   i1 = 0,     i2 = 0;
            float* wout = out + (size_t)4 * n_tokens + (size_t)t * NEXP;
            #pragma unroll 8
            for (int e = 0; e < NEXP; ++e) {
                float p = __expf(lr[e] + gb[e] - mx) * inv;
                wout[e] = p;
                if (p > w1)      { w2 = w1; i2 = i1; w1 = p; i1 = e; }
                else if (p > w2) { w2 = p;  i2 = e; }
            }

            out[(size_t)t * 2 + 0] = w1;
            out[(size_t)t * 2 + 1] = w2;
            float* iout = out + (size_t)2 * n_tokens;
            iout[(size_t)t * 2 + 0] = (float)i1;
            iout[(size_t)t * 2 + 1] = (float)i2;
        }
    }
}

extern "C" void kernel_launch(void* const* d_in, const int* in_sizes, int n_in,
                              void* d_out, int out_size, void* d_ws, size_t ws_size,
                              hipStream_t stream) {
    (void)n_in; (void)out_size; (void)d_ws; (void)ws_size;
    const float* x  = (const float*)d_in[0];
    const float* gw = (const float*)d_in[1];
    const float* gb = (const float*)d_in[2];
    float* out = (float*)d_out;

    const int n_tokens = in_sizes[0] / D_MODEL;     // 16384
    const int blocks   = (n_tokens + TPB - 1) / TPB; // 128

    hipLaunchKernelGGL(gating_fused, dim3(blocks), dim3(256), 0, stream,
                       x, gw, gb, out, n_tokens);
}